// MultiHeadsGATLayer_2680059593212
// MI455X (gfx1250) — compile-verified
//
#include <hip/hip_runtime.h>

// ---------------- problem constants (match reference) ----------------
#define N_NODES 8192
#define S_DIM   128
#define O_DIM   64
#define H_HEADS 4
#define E_EDGES 262144
#define ALPHA   0.2f
#define INV_N   (1.0f / 8192.0f)
#define CAP     480            // max edges per column handled (Poisson(32): P(>480) ~ 0)

typedef __attribute__((ext_vector_type(2))) float    v2f;
typedef __attribute__((ext_vector_type(8))) float    v8f;
typedef __attribute__((ext_vector_type(4))) float    v4f;
typedef __attribute__((ext_vector_type(4))) unsigned v4u;

#if __has_builtin(__builtin_amdgcn_wmma_f32_16x16x4_f32)
#define HAVE_WMMA_F32K4 1
#endif

// ---------------------------------------------------------------------
// Fold W_heads and a_heads into one 128x16 matrix Wc:
//   col h   (h<4): w_src[h][s] = sum_o W[h][s][o] * a[h][o]
//   col 4+h      : w_dst[h][s] = sum_o W[h][s][o] * a[h][64+o]
//   cols 8..15   : zero padding (WMMA tile width)
__global__ void k_weights(const float* __restrict__ W, const float* __restrict__ a,
                          float* __restrict__ Wc) {
  int t = blockIdx.x * 256 + threadIdx.x;
  if (t >= S_DIM * 16) return;
  int s = t >> 4, c = t & 15;
  float acc = 0.f;
  if (c < 8) {
    int h = c & 3;
    const float* wp = W + ((size_t)h * S_DIM + s) * O_DIM;
    const float* ap = a + h * (2 * O_DIM) + ((c >> 2) ? O_DIM : 0);
    #pragma unroll 8
    for (int o = 0; o < O_DIM; ++o) acc += wp[o] * ap[o];
  }
  Wc[s * 16 + c] = acc;
}

__global__ void k_zero(unsigned* __restrict__ p, int n) {
  int t = blockIdx.x * 256 + threadIdx.x;
  if (t < n) p[t] = 0u;
}

// ---------------------------------------------------------------------
// scores[n][0..3] = s_src, scores[n][4..7] = s_dst  via  x(8192x128) @ Wc(128x16)
// One wave per 16-row tile, V_WMMA_F32_16X16X4_F32 accumulating over K=128.
__global__ void __launch_bounds__(256)
k_scores(const float* __restrict__ x, const float* __restrict__ Wc,
         float* __restrict__ scores) {
  int wave = (blockIdx.x * 256 + (int)threadIdx.x) >> 5;
  int lane = threadIdx.x & 31;
  int row0 = wave * 16;
  int m  = lane & 15;       // A: row / B,D: column
  int kh = lane >> 4;       // K half within the 16x4 step
  const float* xr = x + (size_t)(row0 + m) * S_DIM + 2 * kh;
  const float* wr = Wc + (2 * kh) * 16 + m;
  v8f acc = {};
#if defined(HAVE_WMMA_F32K4)
  #pragma unroll 4
  for (int kb = 0; kb < S_DIM; kb += 4) {
    // A 16x4 f32: vgpr0 = K(+0 | +2), vgpr1 = K(+1 | +3) for lanes 0-15 | 16-31
    v2f av; av.x = xr[kb];      av.y = xr[kb + 1];
    // B 4x16 f32: same striping, N = lane&15
    v2f bv; bv.x = wr[kb * 16]; bv.y = wr[kb * 16 + 16];
    acc = __builtin_amdgcn_wmma_f32_16x16x4_f32(false, av, false, bv,
                                                (short)0, acc, false, false);
  }
#else
  // scalar fallback producing the same D layout
  #pragma unroll
  for (int r = 0; r < 8; ++r) {
    const float* xp = x + (size_t)(row0 + kh * 8 + r) * S_DIM;
    float s = 0.f;
    for (int kk = 0; kk < S_DIM; ++kk) s += xp[kk] * Wc[kk * 16 + m];
    acc[r] = s;
  }
#endif
  // D layout: vgpr r -> M = r + 8*(lane>>4), N = lane&15
  int mb = kh * 8;
  #pragma unroll
  for (int r = 0; r < 8; ++r)
    scores[(size_t)(row0 + mb + r) * 16 + m] = acc[r];
}

// ---------------------------------------------------------------------
// Per-edge: leakyReLU logits per head, combine with lin_w/lin_b, scale by values.
// Also builds the per-column (e1) histogram for the CSC.
__global__ void k_edges(const int* __restrict__ edges, const float* __restrict__ values,
                        const float* __restrict__ scores, const float* __restrict__ lin_w,
                        const float* __restrict__ lin_b,
                        float* __restrict__ edge_val, unsigned* __restrict__ col_cnt) {
  int e = blockIdx.x * 256 + threadIdx.x;
  if (e >= E_EDGES) return;
  int i = edges[e];
  int j = edges[E_EDGES + e];
  float mt = lin_b[0];
  #pragma unroll
  for (int h = 0; h < H_HEADS; ++h) {
    float v = scores[(size_t)i * 16 + h] + scores[(size_t)j * 16 + 4 + h];
    v = (v > 0.f) ? v : ALPHA * v;
    mt += v * lin_w[h];
  }
  edge_val[e] = values[e] * mt;
  atomicAdd(&col_cnt[j], 1u);
}

// Exclusive prefix sum of the 8192 column counts (single block).
__global__ void __launch_bounds__(256)
k_scan(const unsigned* __restrict__ col_cnt, unsigned* __restrict__ col_ptr) {
  __shared__ unsigned partial[256];
  __shared__ unsigned offs[257];
  int t = threadIdx.x;
  unsigned local[32];
  unsigned s = 0;
  #pragma unroll
  for (int k = 0; k < 32; ++k) { local[k] = s; s += col_cnt[t * 32 + k]; }
  partial[t] = s;
  __syncthreads();
  if (t == 0) {
    unsigned acc = 0;
    for (int k = 0; k < 256; ++k) { offs[k] = acc; acc += partial[k]; }
    offs[256] = acc;
  }
  __syncthreads();
  unsigned base = offs[t];
  #pragma unroll
  for (int k = 0; k < 32; ++k) col_ptr[t * 32 + k] = base + local[k];
  if (t == 255) col_ptr[N_NODES] = offs[256];
}

__global__ void k_csc(const int* __restrict__ edges, const unsigned* __restrict__ col_ptr,
                      unsigned* __restrict__ col_fill, unsigned* __restrict__ col_eidx) {
  int e = blockIdx.x * 256 + threadIdx.x;
  if (e >= E_EDGES) return;
  int j = edges[E_EDGES + e];
  unsigned pos = col_ptr[j] + atomicAdd(&col_fill[j], 1u);
  col_eidx[pos] = (unsigned)e;
}

// ---------------------------------------------------------------------
// One wave per column: dedup duplicate (row,col) cells (reference sums dups
// and sets mask = count when count >= 2, mask = 0 when count == 1), then
// column max and sum(exp).  Non-edge entries contribute exp(-1e9 - m) == 0.
__global__ void __launch_bounds__(256)
k_colstats(const int* __restrict__ edges, const unsigned* __restrict__ col_ptr,
           const unsigned* __restrict__ col_eidx, const float* __restrict__ edge_val,
           float* __restrict__ edge_vv, float* __restrict__ colmax,
           float* __restrict__ colsum) {
  __shared__ unsigned s_row[8][CAP];
  __shared__ float    s_val[8][CAP];
  __shared__ unsigned s_eid[8][CAP];
  __shared__ float    s_v[8][CAP];
  int w = threadIdx.x >> 5;
  int lane = threadIdx.x & 31;
  int j = blockIdx.x * 8 + w;
  unsigned p0 = col_ptr[j];
  int k = (int)(col_ptr[j + 1] - p0);
  if (k > CAP) k = CAP;
  for (int t = lane; t < k; t += 32) {
    unsigned e = col_eidx[p0 + t];
    s_eid[w][t] = e;
    s_row[w][t] = (unsigned)edges[e];
    s_val[w][t] = edge_val[e];
  }
  __syncthreads();
  float lmax = -3.0e38f;
  for (int t = lane; t < k; t += 32) {
    unsigned row = s_row[w][t];
    int first = CAP, cnt = 0;
    float sum = 0.f;
    for (int u = 0; u < k; ++u) {
      if (s_row[w][u] == row) {
        if (u < first) first = u;
        ++cnt;
        sum += s_val[w][u];
      }
    }
    float v = sum + ((cnt == 1) ? 0.f : (float)cnt);   // dense + mask
    edge_vv[s_eid[w][t]] = v;                          // dup edges get same v
    s_v[w][t] = (first == t) ? v : -3.0e38f;           // leaders only for sum
    lmax = fmaxf(lmax, v);
  }
  #pragma unroll
  for (int d = 16; d > 0; d >>= 1) lmax = fmaxf(lmax, __shfl_xor(lmax, d, 32));
  float lsum = 0.f;
  for (int t = lane; t < k; t += 32) lsum += __expf(s_v[w][t] - lmax);
  #pragma unroll
  for (int d = 16; d > 0; d >>= 1) lsum += __shfl_xor(lsum, d, 32);
  if (lane == 0) { colmax[j] = lmax; colsum[j] = lsum; }
}

// ---------------------------------------------------------------------
// Background: 0 for columns with edges, 1/8192 for empty columns (pure -1e9
// column softmaxes to uniform).  256 MB of coalesced b128 stores -> the
// bandwidth floor of this whole problem.  Non-temporal stores: zero reuse,
// and 256 MB > 192 MB L2, so keep the stream out of the cache hierarchy.
// NOTE: ext-vector types (not HIP float4) — __builtin_nontemporal_store
// only accepts scalar/clang-vector pointees.
#define FILL_BLOCKS 16384
__global__ void __launch_bounds__(256)
k_fill(v4f* __restrict__ out, const v4u* __restrict__ col_cnt4) {
  const size_t total  = (size_t)N_NODES * N_NODES / 4;   // v4f count
  const size_t stride = (size_t)FILL_BLOCKS * 256;
  size_t idx = (size_t)blockIdx.x * 256 + threadIdx.x;
  #pragma unroll 4
  for (; idx < total; idx += stride) {
    int jq = (int)(idx & 2047);                          // 2048 v4f per row
    v4u c = col_cnt4[jq];                                // small, L0/L2 resident
    v4f v;
    v.x = c.x ? 0.f : INV_N;
    v.y = c.y ? 0.f : INV_N;
    v.z = c.z ? 0.f : INV_N;
    v.w = c.w ? 0.f : INV_N;
    __builtin_nontemporal_store(v, &out[idx]);
  }
}

__global__ void k_scatter(const int* __restrict__ edges, const float* __restrict__ edge_vv,
                          const float* __restrict__ colmax, const float* __restrict__ colsum,
                          float* __restrict__ out) {
  int e = blockIdx.x * 256 + threadIdx.x;
  if (e >= E_EDGES) return;
  int i = edges[e];
  int j = edges[E_EDGES + e];
  float r = __expf(edge_vv[e] - colmax[j]) / colsum[j];
  __builtin_nontemporal_store(r, &out[(size_t)i * N_NODES + j]);
}

// ---------------------------------------------------------------------
extern "C" void kernel_launch(void* const* d_in, const int* in_sizes, int n_in,
                              void* d_out, int out_size, void* d_ws, size_t ws_size,
                              hipStream_t stream) {
  (void)in_sizes; (void)n_in; (void)out_size; (void)ws_size;
  const float* x      = (const float*)d_in[0];
  const int*   edges  = (const int*)d_in[1];
  const float* values = (const float*)d_in[2];
  const float* W      = (const float*)d_in[3];
  const float* a      = (const float*)d_in[4];
  const float* lin_w  = (const float*)d_in[5];
  const float* lin_b  = (const float*)d_in[6];
  float* out = (float*)d_out;

  char* ws = (char*)d_ws;
  size_t off = 0;
  auto take = [&](size_t bytes) -> char* {
    char* p = ws + off;
    off += (bytes + 255) & ~(size_t)255;
    return p;
  };
  float*    Wc       = (float*)take((size_t)S_DIM * 16 * 4);
  float*    scores   = (float*)take((size_t)N_NODES * 16 * 4);
  float*    edge_val = (float*)take((size_t)E_EDGES * 4);
  float*    edge_vv  = (float*)take((size_t)E_EDGES * 4);
  unsigned* col_cnt  = (unsigned*)take((size_t)N_NODES * 4);
  unsigned* col_ptr  = (unsigned*)take((size_t)(N_NODES + 1) * 4);
  unsigned* col_fill = (unsigned*)take((size_t)N_NODES * 4);
  unsigned* col_eidx = (unsigned*)take((size_t)E_EDGES * 4);
  float*    colmax   = (float*)take((size_t)N_NODES * 4);
  float*    colsum   = (float*)take((size_t)N_NODES * 4);

  k_weights<<<(S_DIM * 16 + 255) / 256, 256, 0, stream>>>(W, a, Wc);
  k_zero<<<(N_NODES + 255) / 256, 256, 0, stream>>>(col_cnt, N_NODES);
  k_zero<<<(N_NODES + 255) / 256, 256, 0, stream>>>(col_fill, N_NODES);
  // 512 row-tiles, one wave each -> 64 blocks of 8 waves
  k_scores<<<(N_NODES / 16) * 32 / 256, 256, 0, stream>>>(x, Wc, scores);
  k_edges<<<E_EDGES / 256, 256, 0, stream>>>(edges, values, scores, lin_w, lin_b,
                                             edge_val, col_cnt);
  k_scan<<<1, 256, 0, stream>>>(col_cnt, col_ptr);
  k_csc<<<E_EDGES / 256, 256, 0, stream>>>(edges, col_ptr, col_fill, col_eidx);
  k_colstats<<<N_NODES / 8, 256, 0, stream>>>(edges, col_ptr, col_eidx, edge_val,
                                              edge_vv, colmax, colsum);
  k_fill<<<FILL_BLOCKS, 256, 0, stream>>>((v4f*)out, (const v4u*)col_cnt);
  k_scatter<<<E_EDGES / 256, 256, 0, stream>>>(edges, edge_vv, colmax, colsum, out);
}